// HNMCrossEntropyLoss_62646392979575
// MI455X (gfx1250) — compile-verified
//
#include <hip/hip_runtime.h>
#include <hip/hip_bf16.h>
#include <stdint.h>

// Problem constants (fixed by the reference setup)
#define BB   8
#define CC   64
#define NN   131072
#define TOPK 26214        // int(0.2 * 131072)
#define BLK  256
#define NBLK (NN / BLK)   // 512 partial sums per row

typedef __attribute__((ext_vector_type(2))) float v2f;
typedef __attribute__((ext_vector_type(8))) float v8f;

// ---------------------------------------------------------------------------
// init: zero radix histograms, set per-row select state (prefix=0, k=TOPK)
// ---------------------------------------------------------------------------
__global__ void k_init(unsigned* __restrict__ hist,
                       unsigned* __restrict__ stateP,
                       unsigned* __restrict__ stateK) {
  int t = blockIdx.x * blockDim.x + threadIdx.x;
  if (t < BB * 256) hist[t] = 0u;
  if (t < BB) { stateP[t] = 0u; stateK[t] = (unsigned)TOPK; }
}

// ---------------------------------------------------------------------------
// loss: single streaming pass over x (268 MB -> bandwidth bound).
// Each thread owns one n: cache 64 channel values in VGPRs, online max,
// pick x[b,y,n] with a compare-select (no scattered gather), then
// l = max + log(sum exp(v-max)) - x_y, clamped to >= 0 so that the uint bit
// pattern is monotone for radix select.
// ---------------------------------------------------------------------------
__global__ void k_loss(const float* __restrict__ x,
                       const long long* __restrict__ y,
                       float* __restrict__ l) {
  int n = blockIdx.x * BLK + threadIdx.x;
  int b = blockIdx.y;
  const float* xb = x + (size_t)b * CC * NN + n;
  int yv = (int)y[(size_t)b * NN + n];

  float r[CC];
  float m = -__builtin_inff();
  float tgt = 0.0f;
#pragma unroll
  for (int c = 0; c < CC; ++c) {
    float v = __builtin_nontemporal_load(&xb[(size_t)c * NN]);  // streaming: NT hint
    r[c] = v;
    m = fmaxf(m, v);
    tgt = (c == yv) ? v : tgt;
  }
  float s = 0.0f;
#pragma unroll
  for (int c = 0; c < CC; ++c) s += __expf(r[c] - m);  // v_exp_f32

  float loss = m + __logf(s) - tgt;                    // >= 0 mathematically
  l[(size_t)b * NN + n] = fmaxf(loss, 0.0f);           // keep in L2 (4 MB)
}

// ---------------------------------------------------------------------------
// radix-select pass: per-row 256-bin histogram of byte (key>>shift)&255 over
// elements whose high bits match the current prefix. Integer atomics only
// (deterministic). Each block covers 2048 elements.
// ---------------------------------------------------------------------------
__global__ void k_hist(const float* __restrict__ l,
                       unsigned* __restrict__ hist,
                       const unsigned* __restrict__ stateP,
                       int shift) {
  __shared__ unsigned sh[256];
  int b = blockIdx.y;
  sh[threadIdx.x] = 0u;
  __syncthreads();

  unsigned long long prefix = (unsigned long long)stateP[b];
  int hs = shift + 8;  // 32..8 ; u64 shift so hs==32 is well-defined
  int base = blockIdx.x * (BLK * 8);
#pragma unroll
  for (int i = 0; i < 8; ++i) {
    int n = base + i * BLK + threadIdx.x;
    unsigned key = __float_as_uint(l[(size_t)b * NN + n]);
    if (((unsigned long long)key >> hs) == prefix)
      atomicAdd(&sh[(key >> shift) & 255u], 1u);
  }
  __syncthreads();
  unsigned c = sh[threadIdx.x];
  if (c) atomicAdd(&hist[b * 256 + threadIdx.x], c);
}

// ---------------------------------------------------------------------------
// scan: per row, walk histogram from high digit down to find the digit where
// the running count crosses k; extend prefix, update k; zero hist for next
// pass. One block per row.
// ---------------------------------------------------------------------------
__global__ void k_scan(unsigned* __restrict__ hist,
                       unsigned* __restrict__ stateP,
                       unsigned* __restrict__ stateK) {
  __shared__ unsigned cnt[256];
  int b = blockIdx.x;
  cnt[threadIdx.x] = hist[b * 256 + threadIdx.x];
  __syncthreads();
  if (threadIdx.x == 0) {
    unsigned k = stateK[b];
    unsigned acc = 0u;
    int chosen = 0;
    for (int d = 255; d >= 0; --d) {
      unsigned c = cnt[d];
      if (acc + c >= k) { chosen = d; break; }
      acc += c;
    }
    stateP[b] = (stateP[b] << 8) | (unsigned)chosen;
    stateK[b] = k - acc;  // elements equal to threshold still needed
  }
  __syncthreads();
  hist[b * 256 + threadIdx.x] = 0u;  // ready for next pass
}

// ---------------------------------------------------------------------------
// sumgt: per-block fixed-order LDS tree sum of values strictly greater than
// the threshold key T (deterministic). One partial per block.
// ---------------------------------------------------------------------------
__global__ void k_sumgt(const float* __restrict__ l,
                        const unsigned* __restrict__ stateP,
                        float* __restrict__ partials) {
  __shared__ float sd[BLK];
  int b = blockIdx.y;
  int n = blockIdx.x * BLK + threadIdx.x;
  unsigned T = stateP[b];
  float v = l[(size_t)b * NN + n];
  sd[threadIdx.x] = (__float_as_uint(v) > T) ? v : 0.0f;
  __syncthreads();
  for (int s = BLK / 2; s > 0; s >>= 1) {
    if (threadIdx.x < s) sd[threadIdx.x] += sd[threadIdx.x + s];
    __syncthreads();
  }
  if (threadIdx.x == 0) partials[b * NBLK + blockIdx.x] = sd[0];
}

// ---------------------------------------------------------------------------
// final: reduce 512 partials/row with V_WMMA_F32_16X16X4_F32 against an
// all-ones B matrix (row sums, chained accumulation). A-layout per ISA:
// A[m][k] -> lane = m + 16*(k>=2), vgpr = k&1. D-layout: lane j<16 holds rows
// 0..7 in vgprs 0..7 (any column j carries the same row sum), lanes>=16 hold
// rows 8..15. Fixed dataflow -> deterministic. Single wave, EXEC all ones.
// ---------------------------------------------------------------------------
__global__ void k_final(const float* __restrict__ partials,
                        const unsigned* __restrict__ stateP,
                        const unsigned* __restrict__ stateK,
                        float* __restrict__ out) {
  int lane = threadIdx.x;        // blockDim.x == 32, one wave
  int m = lane & 15;
  int h = lane >> 4;

  v2f bones;
  bones.x = 1.0f; bones.y = 1.0f;   // all-ones B: layout-invariant

  float grand = 0.0f;
  for (int b = 0; b < BB; ++b) {
    const float* pb = partials + b * NBLK;
    v8f acc = {};
#pragma unroll
    for (int t = 0; t < NBLK / 64; ++t) {  // 8 chained 16x16x4 WMMAs
      int idx = t * 64 + m * 4 + h * 2;
      v2f a;
      a.x = pb[idx];
      a.y = pb[idx + 1];
      acc = __builtin_amdgcn_wmma_f32_16x16x4_f32(
          /*neg_a=*/false, a, /*neg_b=*/false, bones,
          /*c_mod=*/(short)0, acc, /*reuse_a=*/false, /*reuse_b=*/false);
    }
    // lane<16: rows 0..7 ; lane>=16: rows 8..15 (column = lane%16, all equal)
    float rs = acc[0] + acc[1] + acc[2] + acc[3] + acc[4] + acc[5] + acc[6] + acc[7];
    float sum_gt = __shfl(rs, 0, 32) + __shfl(rs, 16, 32);

    float T = __uint_as_float(stateP[b]);     // exact k-th largest value
    float row_top = sum_gt + (float)stateK[b] * T;  // tie-exact top-k sum
    grand += row_top;
  }
  if (lane == 0) out[0] = grand / ((float)TOPK * (float)BB);
}

// ---------------------------------------------------------------------------
extern "C" void kernel_launch(void* const* d_in, const int* in_sizes, int n_in,
                              void* d_out, int out_size, void* d_ws, size_t ws_size,
                              hipStream_t stream) {
  const float* x = (const float*)d_in[0];          // [8,64,131072] f32
  const long long* y = (const long long*)d_in[1];  // [8,131072] i64
  float* out = (float*)d_out;                      // scalar f32

  char* ws = (char*)d_ws;
  float*    l        = (float*)   (ws);                          // 4 MB
  unsigned* hist     = (unsigned*)(ws + (size_t)BB * NN * 4);    // 8 KB
  unsigned* stateP   = hist + BB * 256;                          // 32 B
  unsigned* stateK   = stateP + BB;                              // 32 B
  float*    partials = (float*)(stateK + BB);                    // 16 KB

  k_init<<<dim3((BB * 256 + 255) / 256), 256, 0, stream>>>(hist, stateP, stateK);
  k_loss<<<dim3(NN / BLK, BB), BLK, 0, stream>>>(x, y, l);

  // 4 radix passes: shift = 24, 16, 8, 0
  for (int p = 0; p < 4; ++p) {
    int shift = 24 - 8 * p;
    k_hist<<<dim3(NN / (BLK * 8), BB), BLK, 0, stream>>>(l, hist, stateP, shift);
    k_scan<<<dim3(BB), 256, 0, stream>>>(hist, stateP, stateK);
  }

  k_sumgt<<<dim3(NN / BLK, BB), BLK, 0, stream>>>(l, stateP, partials);
  k_final<<<1, 32, 0, stream>>>(partials, stateP, stateK, out);
}